// AttentionDecoder_61967788146774
// MI455X (gfx1250) — compile-verified
//
#include <hip/hip_runtime.h>
#include <hip/hip_bf16.h>
#include <math.h>

#define B_   64
#define S_   400
#define SX_  200
#define H_   512
#define E_   256
#define V_   50000
#define EXT_ 1000
#define H2_  (2*H_)

#define KP_   64     // K panel staged in LDS
#define APAD_ 68     // padded LDS row stride (floats): 4r+kb covers all 64 banks

typedef __attribute__((ext_vector_type(2))) float v2f;
typedef __attribute__((ext_vector_type(8))) float v8f;

__device__ __forceinline__ float wave_reduce_sum(float v) {
  #pragma unroll
  for (int off = 16; off > 0; off >>= 1) v += __shfl_xor(v, off, 32);
  return v;
}

// ---------------------------------------------------------------------------
// Build lstm_in = [emb[y], c_t_1]  (64 x 1280)
// ---------------------------------------------------------------------------
__global__ void prep_lstm_in(const int* __restrict__ y, const float* __restrict__ emb,
                             const float* __restrict__ c_t_1, float* __restrict__ lstm_in) {
  int idx = blockIdx.x * blockDim.x + threadIdx.x;
  const int W = E_ + H2_;
  if (idx >= B_ * W) return;
  int b = idx / W, j = idx % W;
  float v;
  if (j < E_) v = emb[(long)y[b] * E_ + j];
  else        v = c_t_1[b * H2_ + (j - E_)];
  lstm_in[idx] = v;
}

// ---------------------------------------------------------------------------
// Multi-wave WMMA fp32 GEMM: C[64 x N] (+)= A[64 x K] * Bt[N x K]^T + bias[N]
// 256 threads = 8 waves; block covers 128 N columns; A K-panels staged in LDS
// and shared by all 8 waves (8x less A traffic). B fragments batch-loaded per
// panel; next B panel prefetched (global_prefetch_b8). V_WMMA_F32_16X16X4_F32.
// ---------------------------------------------------------------------------
__global__ __launch_bounds__(256) void wmma_gemm_f32_mw(
    const float* __restrict__ A, const float* __restrict__ Bt,
    const float* __restrict__ bias, float* __restrict__ C,
    int N, int K, int acc)
{
  __shared__ float As[64 * APAD_];
  const int tid    = threadIdx.x;
  const int lane   = tid & 31;
  const int wv     = tid >> 5;                    // wave id 0..7
  const int n0     = blockIdx.x * 128 + wv * 16;  // wave's 16-col tile
  const bool active = (n0 < N);                   // wave-uniform; EXEC stays all-1s
  const int col    = n0 + (lane & 15);
  const int kh     = (lane >> 4) << 1;            // 0 or 2
  const int r0     = lane & 15;
  const float* Brow = Bt + (long)(active ? col : 0) * K;

  v8f c0 = {}, c1 = {}, c2 = {}, c3 = {};

  for (int kp = 0; kp < K; kp += KP_) {
    // ---- cooperative A panel load: 64 x 64 floats = 1024 float4 slots ----
    for (int s = tid; s < 64 * (KP_ / 4); s += 256) {
      int row = s >> 4;          // 16 float4 slots per row
      int c4  = (s & 15) << 2;
      float4 vsrc = *(const float4*)(A + (long)row * K + kp + c4);
      *(float4*)(&As[row * APAD_ + c4]) = vsrc;
    }
    __syncthreads();

    if (active) {
      if (kp + KP_ < K) __builtin_prefetch(Brow + kp + KP_, 0, 0);
      // ---- batch-load all B fragments for this panel ----
      v2f bf[KP_ / 4];
      #pragma unroll
      for (int j = 0; j < KP_ / 4; ++j) {
        int kb = kp + 4 * j + kh;
        bf[j].x = Brow[kb];
        bf[j].y = Brow[kb + 1];
      }
      // ---- WMMA burst: A fragments from LDS (ds_load_b64, bank-padded) ----
      #pragma unroll
      for (int j = 0; j < KP_ / 4; ++j) {
        const float* as = &As[4 * j + kh];
        v2f a0, a1, a2, a3;
        a0.x = as[(r0 +  0) * APAD_]; a0.y = as[(r0 +  0) * APAD_ + 1];
        a1.x = as[(r0 + 16) * APAD_]; a1.y = as[(r0 + 16) * APAD_ + 1];
        a2.x = as[(r0 + 32) * APAD_]; a2.y = as[(r0 + 32) * APAD_ + 1];
        a3.x = as[(r0 + 48) * APAD_]; a3.y = as[(r0 + 48) * APAD_ + 1];
        c0 = __builtin_amdgcn_wmma_f32_16x16x4_f32(false, a0, false, bf[j], (short)0, c0, false, false);
        c1 = __builtin_amdgcn_wmma_f32_16x16x4_f32(false, a1, false, bf[j], (short)0, c1, false, false);
        c2 = __builtin_amdgcn_wmma_f32_16x16x4_f32(false, a2, false, bf[j], (short)0, c2, false, false);
        c3 = __builtin_amdgcn_wmma_f32_16x16x4_f32(false, a3, false, bf[j], (short)0, c3, false, false);
      }
    }
    __syncthreads();
  }

  if (active) {
    const float bv   = bias ? bias[col] : 0.0f;
    const int   rofs = (lane >> 4) * 8;
    #pragma unroll
    for (int i = 0; i < 8; ++i) {
      long idx;
      idx = (long)( 0 + i + rofs) * N + col; C[idx] = (acc ? C[idx] : 0.0f) + c0[i] + bv;
      idx = (long)(16 + i + rofs) * N + col; C[idx] = (acc ? C[idx] : 0.0f) + c1[i] + bv;
      idx = (long)(32 + i + rofs) * N + col; C[idx] = (acc ? C[idx] : 0.0f) + c2[i] + bv;
      idx = (long)(48 + i + rofs) * N + col; C[idx] = (acc ? C[idx] : 0.0f) + c3[i] + bv;
    }
  }
}

// ---------------------------------------------------------------------------
// LSTM elementwise: c_new/h_new; writes s_t_cat=[h|c], dec_out[:, :H]=h, d_out h/c
// ---------------------------------------------------------------------------
__global__ void lstm_elem(const float* __restrict__ gates, const float* __restrict__ c0s,
                          float* __restrict__ s_t_cat, float* __restrict__ dec_out,
                          float* __restrict__ out_h, float* __restrict__ out_c) {
  int idx = blockIdx.x * blockDim.x + threadIdx.x;
  if (idx >= B_ * H_) return;
  int b = idx / H_, j = idx % H_;
  const float* g = gates + (long)b * 4 * H_;
  float gi = g[j], gf = g[H_ + j], gg = g[2 * H_ + j], go = g[3 * H_ + j];
  float si = 1.f / (1.f + __expf(-gi));
  float sf = 1.f / (1.f + __expf(-gf));
  float so = 1.f / (1.f + __expf(-go));
  float cn = sf * c0s[idx] + si * tanhf(gg);
  float hn = so * tanhf(cn);
  s_t_cat[b * H2_ + j]       = hn;
  s_t_cat[b * H2_ + H_ + j]  = cn;
  dec_out[b * (H_ + H2_) + j] = hn;
  out_h[idx] = hn;
  out_c[idx] = cn;
}

// ---------------------------------------------------------------------------
// e[b,l] = sum_d tanh(feat[b,l,d] + s_ws[b,d]) * v[d]   (one wave per (b,l))
// ---------------------------------------------------------------------------
__global__ void attn_scores(const float* __restrict__ feat, const float* __restrict__ s_ws,
                            const float* __restrict__ vw, float* __restrict__ e, int L) {
  int wave = (blockIdx.x * blockDim.x + threadIdx.x) >> 5;
  int lane = threadIdx.x & 31;
  if (wave >= B_ * L) return;
  int b = wave / L, l = wave % L;
  const float* f = feat + ((long)b * L + l) * H2_;
  const float* s = s_ws + (long)b * H2_;
  float acc = 0.f;
  for (int d = lane; d < H2_; d += 32) acc += tanhf(f[d] + s[d]) * vw[d];
  acc = wave_reduce_sum(acc);
  if (lane == 0) e[(long)b * L + l] = acc;
}

// ---------------------------------------------------------------------------
// softmax over L, * mask, renorm by masked sum (+eps); in-place; opt copy out
// ---------------------------------------------------------------------------
__global__ void attn_softmax(float* __restrict__ e, const float* __restrict__ mask,
                             float* __restrict__ attn_copy, int L) {
  __shared__ float red[512];
  int b = blockIdx.x, t = threadIdx.x;
  float x = (t < L) ? e[(long)b * L + t] : -INFINITY;
  red[t] = x; __syncthreads();
  for (int s = 256; s > 0; s >>= 1) { if (t < s) red[t] = fmaxf(red[t], red[t + s]); __syncthreads(); }
  float m = red[0]; __syncthreads();
  float p = (t < L) ? __expf(x - m) : 0.f;
  red[t] = p; __syncthreads();
  for (int s = 256; s > 0; s >>= 1) { if (t < s) red[t] += red[t + s]; __syncthreads(); }
  float sum = red[0]; __syncthreads();
  float a = (t < L) ? (p / sum) * mask[(long)b * L + t] : 0.f;
  red[t] = a; __syncthreads();
  for (int s = 256; s > 0; s >>= 1) { if (t < s) red[t] += red[t + s]; __syncthreads(); }
  float s2 = red[0];
  if (t < L) {
    float att = a / (s2 + 1e-12f);
    e[(long)b * L + t] = att;
    if (attn_copy) attn_copy[(long)b * L + t] = att;
  }
}

// ---------------------------------------------------------------------------
// c[b,d] = sum_l attn[b,l] * outs[b,l,d]
// ---------------------------------------------------------------------------
__global__ void context_kernel(const float* __restrict__ attn, const float* __restrict__ outs,
                               float* __restrict__ c, int L) {
  int idx = blockIdx.x * blockDim.x + threadIdx.x;
  if (idx >= B_ * H2_) return;
  int b = idx / H2_, d = idx % H2_;
  const float* o = outs + (long)b * L * H2_ + d;
  const float* a = attn + (long)b * L;
  float acc = 0.f;
  for (int l = 0; l < L; ++l) acc += a[l] * o[(long)l * H2_];
  c[idx] = acc;
}

// ---------------------------------------------------------------------------
// c_t = log(exp(ec)+exp(xc)); write dec_out[:, H:] and d_out c_t slot
// ---------------------------------------------------------------------------
__global__ void logadd(const float* __restrict__ ec, const float* __restrict__ xc,
                       float* __restrict__ dec_out, float* __restrict__ out_ct) {
  int idx = blockIdx.x * blockDim.x + threadIdx.x;
  if (idx >= B_ * H2_) return;
  int b = idx / H2_, j = idx % H2_;
  float x = ec[idx], y = xc[idx];
  float m = fmaxf(x, y);
  float r = m + __logf(__expf(x - m) + __expf(y - m));
  dec_out[b * (H_ + H2_) + H_ + j] = r;
  out_ct[idx] = r;
}

// ---------------------------------------------------------------------------
// row softmax over V=50000, in place
// ---------------------------------------------------------------------------
__global__ void vocab_softmax(float* __restrict__ logits) {
  __shared__ float red[1024];
  int b = blockIdx.x, t = threadIdx.x;
  float* row = logits + (long)b * V_;
  float m = -INFINITY;
  for (int v = t; v < V_; v += 1024) m = fmaxf(m, row[v]);
  red[t] = m; __syncthreads();
  for (int s = 512; s > 0; s >>= 1) { if (t < s) red[t] = fmaxf(red[t], red[t + s]); __syncthreads(); }
  m = red[0]; __syncthreads();
  float sum = 0.f;
  for (int v = t; v < V_; v += 1024) sum += __expf(row[v] - m);
  red[t] = sum; __syncthreads();
  for (int s = 512; s > 0; s >>= 1) { if (t < s) red[t] += red[t + s]; __syncthreads(); }
  float inv = 1.f / red[0];
  for (int v = t; v < V_; v += 1024) row[v] = __expf(row[v] - m) * inv;
}

// ---------------------------------------------------------------------------
// p_gen[b] = sigmoid([c_t | s_t_cat | dec_emb] . pg_W + pg_b); one wave per b
// ---------------------------------------------------------------------------
__global__ __launch_bounds__(32) void pgen_kernel(
    const float* __restrict__ dec_out, const float* __restrict__ s_t_cat,
    const float* __restrict__ lstm_in, const float* __restrict__ pg_W,
    const float* __restrict__ pg_b, float* __restrict__ p_gen) {
  int b = blockIdx.x, lane = threadIdx.x & 31;
  float acc = 0.f;
  const float* ct = dec_out + b * (H_ + H2_) + H_;
  for (int j = lane; j < H2_; j += 32) acc += ct[j] * pg_W[j];
  const float* st = s_t_cat + (long)b * H2_;
  for (int j = lane; j < H2_; j += 32) acc += st[j] * pg_W[H2_ + j];
  const float* de = lstm_in + (long)b * (E_ + H2_);
  for (int j = lane; j < E_; j += 32) acc += de[j] * pg_W[2 * H2_ + j];
  acc = wave_reduce_sum(acc);
  if (lane == 0) p_gen[b] = 1.f / (1.f + __expf(-(acc + pg_b[0])));
}

// ---------------------------------------------------------------------------
// final[b,v] = p_gen*vocab (v<V) else 0
// ---------------------------------------------------------------------------
__global__ void final_base(const float* __restrict__ vocab, const float* __restrict__ p_gen,
                           float* __restrict__ out) {
  long idx = (long)blockIdx.x * blockDim.x + threadIdx.x;
  const long total = (long)B_ * (V_ + EXT_);
  if (idx >= total) return;
  int b = (int)(idx / (V_ + EXT_));
  int v = (int)(idx % (V_ + EXT_));
  out[idx] = (v < V_) ? p_gen[b] * vocab[(long)b * V_ + v] : 0.f;
}

// ---------------------------------------------------------------------------
// scatter-add (1-p_gen)*attn at enc_inps_extended
// ---------------------------------------------------------------------------
__global__ void scatter_add(const float* __restrict__ attn, const float* __restrict__ p_gen,
                            const int* __restrict__ inps, float* __restrict__ out) {
  int idx = blockIdx.x * blockDim.x + threadIdx.x;
  if (idx >= B_ * S_) return;
  int b = idx / S_;
  float w = (1.f - p_gen[b]) * attn[idx];
  atomicAdd(&out[(long)b * (V_ + EXT_) + inps[idx]], w);
}

// ---------------------------------------------------------------------------
extern "C" void kernel_launch(void* const* d_in, const int* in_sizes, int n_in,
                              void* d_out, int out_size, void* d_ws, size_t ws_size,
                              hipStream_t stream) {
  (void)in_sizes; (void)n_in; (void)out_size; (void)ws_size;
  const int*   y_t_1    = (const int*)  d_in[0];
  const float* h_t_1    = (const float*)d_in[1];
  const float* c_state1 = (const float*)d_in[2];
  const float* c_t_1    = (const float*)d_in[3];
  const float* enc_out  = (const float*)d_in[4];
  const float* enc_feat = (const float*)d_in[5];
  const float* enc_mask = (const float*)d_in[6];
  const float* xem_out  = (const float*)d_in[7];
  const float* xem_feat = (const float*)d_in[8];
  const float* xem_mask = (const float*)d_in[9];
  const int*   enc_inps = (const int*)  d_in[11];
  const float* emb      = (const float*)d_in[12];
  const float* Wxc      = (const float*)d_in[13];
  const float* bxc      = (const float*)d_in[14];
  const float* W_ih     = (const float*)d_in[15];
  const float* W_hh     = (const float*)d_in[16];
  const float* b_ih     = (const float*)d_in[17];
  const float* b_hh     = (const float*)d_in[18];
  const float* Ws_W     = (const float*)d_in[19];
  const float* Ws_b     = (const float*)d_in[20];
  const float* v_w      = (const float*)d_in[21];
  const float* V1_W     = (const float*)d_in[22];
  const float* V1_b     = (const float*)d_in[23];
  const float* V2_W     = (const float*)d_in[24];
  const float* V2_b     = (const float*)d_in[25];
  const float* pg_W     = (const float*)d_in[26];
  const float* pg_b     = (const float*)d_in[27];

  // workspace layout (floats)
  float* ws = (float*)d_ws;
  float* lstm_in = ws;                      // 64*1280
  float* x       = lstm_in + B_*(E_+H2_);   // 64*256
  float* gates   = x       + B_*E_;         // 64*2048
  float* s_t_cat = gates   + B_*4*H_;       // 64*1024
  float* dec_out = s_t_cat + B_*H2_;        // 64*1536
  float* s_ws    = dec_out + B_*(H_+H2_);   // 64*1024
  float* e_enc   = s_ws    + B_*H2_;        // 64*400
  float* e_xem   = e_enc   + B_*S_;         // 64*200
  float* enc_c   = e_xem   + B_*SX_;        // 64*1024
  float* xem_c   = enc_c   + B_*H2_;        // 64*1024
  float* tmp1    = xem_c   + B_*H2_;        // 64*512
  float* p_gen   = tmp1    + B_*H_;         // 64
  float* logits  = p_gen   + B_;            // 64*50000

  // output layout (floats)
  float* out        = (float*)d_out;
  float* out_final  = out;                                    // 64*51000
  float* out_h      = out_final + (long)B_*(V_+EXT_);         // 64*512
  float* out_c      = out_h + B_*H_;                          // 64*512
  float* out_ct     = out_c + B_*H_;                          // 64*1024
  float* out_attn   = out_ct + B_*H2_;                        // 64*400

  #define GEMM_GRID(N) (((N) + 127) / 128)

  // 1. embed gather + concat with c_t_1
  prep_lstm_in<<<(B_*(E_+H2_) + 255)/256, 256, 0, stream>>>(y_t_1, emb, c_t_1, lstm_in);
  // 2. x = lstm_in @ Wxc^T + bxc   (N=256, K=1280)
  wmma_gemm_f32_mw<<<GEMM_GRID(E_), 256, 0, stream>>>(lstm_in, Wxc, bxc, x, E_, E_+H2_, 0);
  // 3. gates = x@W_ih^T + b_ih  then  += h0@W_hh^T + b_hh
  wmma_gemm_f32_mw<<<GEMM_GRID(4*H_), 256, 0, stream>>>(x,     W_ih, b_ih, gates, 4*H_, E_, 0);
  wmma_gemm_f32_mw<<<GEMM_GRID(4*H_), 256, 0, stream>>>(h_t_1, W_hh, b_hh, gates, 4*H_, H_, 1);
  // 4. LSTM cell elementwise
  lstm_elem<<<(B_*H_ + 255)/256, 256, 0, stream>>>(gates, c_state1, s_t_cat, dec_out, out_h, out_c);
  // 5. s_t_ws = s_t_cat @ Ws_W^T + Ws_b  (N=1024, K=1024)
  wmma_gemm_f32_mw<<<GEMM_GRID(H2_), 256, 0, stream>>>(s_t_cat, Ws_W, Ws_b, s_ws, H2_, H2_, 0);
  // 6. attention (enc)
  attn_scores<<<(B_*S_*32 + 255)/256, 256, 0, stream>>>(enc_feat, s_ws, v_w, e_enc, S_);
  attn_softmax<<<B_, 512, 0, stream>>>(e_enc, enc_mask, out_attn, S_);
  context_kernel<<<(B_*H2_ + 255)/256, 256, 0, stream>>>(e_enc, enc_out, enc_c, S_);
  // 7. attention (xem)
  attn_scores<<<(B_*SX_*32 + 255)/256, 256, 0, stream>>>(xem_feat, s_ws, v_w, e_xem, SX_);
  attn_softmax<<<B_, 512, 0, stream>>>(e_xem, xem_mask, (float*)nullptr, SX_);
  context_kernel<<<(B_*H2_ + 255)/256, 256, 0, stream>>>(e_xem, xem_out, xem_c, SX_);
  // 8. c_t = logaddexp(enc_c, xem_c)
  logadd<<<(B_*H2_ + 255)/256, 256, 0, stream>>>(enc_c, xem_c, dec_out, out_ct);
  // 9. vocab: tmp1 = dec_out@V1^T + b  (N=512,K=1536); logits = tmp1@V2^T + b (N=50000,K=512)
  wmma_gemm_f32_mw<<<GEMM_GRID(H_), 256, 0, stream>>>(dec_out, V1_W, V1_b, tmp1, H_, H_+H2_, 0);
  wmma_gemm_f32_mw<<<GEMM_GRID(V_), 256, 0, stream>>>(tmp1, V2_W, V2_b, logits, V_, H_, 0);
  vocab_softmax<<<B_, 1024, 0, stream>>>(logits);
  // 10. p_gen
  pgen_kernel<<<B_, 32, 0, stream>>>(dec_out, s_t_cat, lstm_in, pg_W, pg_b, p_gen);
  // 11. final distribution + scatter add of copy distribution
  final_base<<<(int)(((long)B_*(V_+EXT_) + 255)/256), 256, 0, stream>>>(logits, p_gen, out_final);
  scatter_add<<<(B_*S_ + 255)/256, 256, 0, stream>>>(e_enc, p_gen, enc_inps, out_final);
}